// Low_Rank_Three_Way_RNN_Dynamics_54314156425437
// MI455X (gfx1250) — compile-verified
//
#include <hip/hip_runtime.h>
#include <hip/hip_bf16.h>

#define HID 512
#define RANK 4
#define IN_SIZE 3
#define OUT_SIZE 3
#define BATCH 128
#define SEQ 1000
#define NOISE_STD 0.05f
#define TAU 0.2f

typedef __attribute__((ext_vector_type(2))) float v2f;
typedef __attribute__((ext_vector_type(8))) float v8f;

// tanh via hardware exp2 + hardware rcp (v_exp_f32 / v_rcp_f32, ~1 ulp):
//   tanh(x) = 1 - 2 * rcp(exp2(2*log2e*x) + 1)
// NaN-safe at +/-inf: e->inf gives 1, e->0 gives -1. Avoids the IEEE
// v_div_fmas/v_div_fixup sequence the plain '/' lowers to.
__device__ __forceinline__ float fast_tanh(float x) {
    float e = __builtin_amdgcn_exp2f(x * 2.88539008177792681472f); // 2*log2(e)
    return __builtin_fmaf(-2.0f, __builtin_amdgcn_rcpf(e + 1.0f), 1.0f);
}

// ---------------------------------------------------------------------------
// Kernel 1: sequential scan. One wave32 per batch row (rows are independent,
// so no barriers on the 1000-step critical path). Lane owns h = lane + 32*j,
// j = 0..15 -> fully coalesced noise loads / traj stores. Rank-4 reductions
// done with in-wave butterfly shuffles. Weights staged in LDS once.
// ---------------------------------------------------------------------------
__global__ __launch_bounds__(32) void rnn_scan_kernel(
    const float* __restrict__ u, const float* __restrict__ x0,
    const float* __restrict__ noise, const float* __restrict__ L,
    const float* __restrict__ M, const float* __restrict__ N,
    const float* __restrict__ bias, const float* __restrict__ W_in,
    float* __restrict__ x_final, float* __restrict__ traj)
{
    __shared__ __align__(16) float sM[HID * RANK];
    __shared__ __align__(16) float sN[HID * RANK];
    __shared__ __align__(16) float sL[HID * RANK];
    __shared__ __align__(16) float sWin[HID * IN_SIZE];
    __shared__ __align__(16) float sB[HID];

    const int b    = blockIdx.x;
    const int lane = threadIdx.x;

    for (int i = lane; i < HID * RANK; i += 32) { sM[i] = M[i]; sN[i] = N[i]; sL[i] = L[i]; }
    for (int i = lane; i < HID * IN_SIZE; i += 32) sWin[i] = W_in[i];
    for (int i = lane; i < HID; i += 32) sB[i] = bias[i];
    __syncthreads();  // single-wave workgroup: lowers to s_nop, zero cost

    const float4* sM4 = (const float4*)sM;
    const float4* sN4 = (const float4*)sN;
    const float4* sL4 = (const float4*)sL;

    float x[16], r[16];
    float* trajb = traj + (size_t)b * (SEQ + 1) * HID;

    #pragma unroll
    for (int j = 0; j < 16; ++j) {
        const int h = lane + 32 * j;
        float v = x0[(size_t)b * HID + h];
        x[j] = v;
        r[j] = fast_tanh(v);       // r0 = tanh(x0), no bias (as in reference)
        trajb[h] = v;              // trajectories[:, 0, :]
    }

    const float inv_h2 = 1.0f / ((float)HID * (float)HID);

    for (int t = 0; t < SEQ; ++t) {
        // Uniform (per-block) input read -> scalar loads
        const float* ut = u + ((size_t)b * SEQ + t) * IN_SIZE;
        const float u0 = ut[0], u1 = ut[1], u2 = ut[2];

        const float* nt = noise + ((size_t)t * BATCH + b) * HID;
        if (t + 1 < SEQ) {
            // Prefetch next step's 2KB noise row (64B per lane) to hide HBM
            // latency inside the serial dependency chain.
            __builtin_prefetch(noise + ((size_t)(t + 1) * BATCH + b) * HID + lane * 16, 0, 3);
        }

        // Rank-4 partial reductions rM, rN over this lane's 16 hid elements
        float pm[RANK] = {0.f, 0.f, 0.f, 0.f};
        float pn[RANK] = {0.f, 0.f, 0.f, 0.f};
        #pragma unroll
        for (int j = 0; j < 16; ++j) {
            const int h = lane + 32 * j;
            const float rj = r[j];
            const float4 mv = sM4[h];
            const float4 nv = sN4[h];
            pm[0] += rj * mv.x; pm[1] += rj * mv.y; pm[2] += rj * mv.z; pm[3] += rj * mv.w;
            pn[0] += rj * nv.x; pn[1] += rj * nv.y; pn[2] += rj * nv.z; pn[3] += rj * nv.w;
        }
        // Wave32 butterfly reduction (no barriers, no LDS)
        #pragma unroll
        for (int off = 16; off >= 1; off >>= 1) {
            #pragma unroll
            for (int k = 0; k < RANK; ++k) {
                pm[k] += __shfl_xor(pm[k], off, 32);
                pn[k] += __shfl_xor(pn[k], off, 32);
            }
        }
        const float p0 = pm[0] * pn[0] * inv_h2;
        const float p1 = pm[1] * pn[1] * inv_h2;
        const float p2 = pm[2] * pn[2] * inv_h2;
        const float p3 = pm[3] * pn[3] * inv_h2;

        float* trajt = trajb + (size_t)(t + 1) * HID;
        #pragma unroll
        for (int j = 0; j < 16; ++j) {
            const int h = lane + 32 * j;
            const float4 lv = sL4[h];
            const float hid_w = p0 * lv.x + p1 * lv.y + p2 * lv.z + p3 * lv.w;
            const float inI = u0 * sWin[h * 3 + 0] + u1 * sWin[h * 3 + 1] + u2 * sWin[h * 3 + 2];
            const float nz = nt[h];
            float xv = x[j];
            xv = xv + NOISE_STD * nz + TAU * (-xv + hid_w + inI);
            x[j] = xv;
            trajt[h] = xv;                       // trajectories[:, t+1, :]
            r[j] = fast_tanh(xv + sB[h]);        // r = tanh(x + bias)
        }
    }

    #pragma unroll
    for (int j = 0; j < 16; ++j) {
        const int h = lane + 32 * j;
        x_final[(size_t)b * HID + h] = x[j];
    }
}

// ---------------------------------------------------------------------------
// Kernel 2: output projection out = tanh(traj[:,1:,:]) @ W_out^T as an exact
// fp32 GEMM using V_WMMA_F32_16X16X4_F32 (K=4 native), chained 128x over
// K=512. One wave per 16-row tile of the logical (BATCH*SEQ) x HID matrix.
// A fragment layout (ISA 16x4 f32): lanes 0-15 -> M=lane, VGPR0=K0/VGPR1=K1;
// lanes 16-31 -> M=lane-16, VGPR0=K2/VGPR1=K3. B is W_out^T (4x16 tiles,
// N padded 3->16 with zeros). D layout: lane<16 -> (M=j, N=lane),
// lane>=16 -> (M=8+j, N=lane-16).
// ---------------------------------------------------------------------------
__global__ __launch_bounds__(32) void out_proj_wmma_kernel(
    const float* __restrict__ traj, const float* __restrict__ W_out,
    float* __restrict__ out)
{
    const int tile = blockIdx.x;        // 0 .. BATCH*SEQ/16 - 1
    const int lane = threadIdx.x;
    const int rl   = lane & 15;
    const int hi   = lane >> 4;         // 0: K pair (0,1), 1: K pair (2,3)

    // Per-lane A row pointer: logical row -> (b, t) -> traj[b, t+1, :]
    const int row = tile * 16 + rl;
    const int b   = row / SEQ;
    const int t   = row - b * SEQ;
    const float* arow = traj + ((size_t)b * (SEQ + 1) + (t + 1)) * HID;

    // Per-lane B column: n = rl (zero-masked for n >= OUT_SIZE)
    const int   ncl   = (rl < OUT_SIZE) ? rl : 0;
    const float bmask = (rl < OUT_SIZE) ? 1.0f : 0.0f;
    const float* bcol = W_out + (size_t)ncl * HID;

    v8f acc = {0.f, 0.f, 0.f, 0.f, 0.f, 0.f, 0.f, 0.f};
    for (int k0 = 0; k0 < HID; k0 += 4) {
        const int ka = k0 + 2 * hi;
        v2f a, bb;
        a.x  = fast_tanh(arow[ka]);
        a.y  = fast_tanh(arow[ka + 1]);
        bb.x = bmask * bcol[ka];
        bb.y = bmask * bcol[ka + 1];
        // 8 args: (neg_a, A, neg_b, B, c_mod, C, reuse_a, reuse_b)
        acc = __builtin_amdgcn_wmma_f32_16x16x4_f32(
            false, a, false, bb, (short)0, acc, false, false);
    }

    // Store D columns 0..OUT_SIZE-1
    if (rl < OUT_SIZE) {
        #pragma unroll
        for (int j = 0; j < 8; ++j) {
            const int m    = hi ? (8 + j) : j;
            const int orow = tile * 16 + m;
            out[(size_t)orow * OUT_SIZE + rl] = acc[j];
        }
    }
}

// ---------------------------------------------------------------------------
// d_in order (setup_inputs dict): u, x0, noise, L, M, N, bias_tensor, W_in, W_out
// d_out (flat, fp32): output(128*1000*3) | x_final(128*512) | traj(128*1001*512)
// ---------------------------------------------------------------------------
extern "C" void kernel_launch(void* const* d_in, const int* in_sizes, int n_in,
                              void* d_out, int out_size, void* d_ws, size_t ws_size,
                              hipStream_t stream) {
    const float* u     = (const float*)d_in[0];
    const float* x0    = (const float*)d_in[1];
    const float* noise = (const float*)d_in[2];
    const float* L     = (const float*)d_in[3];
    const float* M     = (const float*)d_in[4];
    const float* N     = (const float*)d_in[5];
    const float* bias  = (const float*)d_in[6];
    const float* W_in  = (const float*)d_in[7];
    const float* W_out = (const float*)d_in[8];

    float* out     = (float*)d_out;
    float* x_final = out + (size_t)BATCH * SEQ * OUT_SIZE;
    float* traj    = x_final + (size_t)BATCH * HID;

    rnn_scan_kernel<<<BATCH, 32, 0, stream>>>(u, x0, noise, L, M, N, bias, W_in,
                                              x_final, traj);
    out_proj_wmma_kernel<<<(BATCH * SEQ) / 16, 32, 0, stream>>>(traj, W_out, out);
}